// LightGCN_38397007626496
// MI455X (gfx1250) — compile-verified
//
#include <hip/hip_runtime.h>

#define DIM 128
#define KEEP_PROB 0.8f
#define INV_KEEP 1.25f

typedef __attribute__((ext_vector_type(2))) float v2f;
typedef __attribute__((ext_vector_type(8))) float v8f;

__device__ __forceinline__ float bcast_f32(float v, int srcLane) {
    return __uint_as_float(__builtin_amdgcn_readlane(__float_as_uint(v), srcLane));
}

// ---------------------------------------------------------------------------
// Zero the workspace (u1, i1, bitmaps) with 128-bit stores.
// ---------------------------------------------------------------------------
__global__ void lgcn_zero_kernel(uint4* __restrict__ ws, long n4) {
    long i = (long)blockIdx.x * blockDim.x + threadIdx.x;
    const long stride = (long)gridDim.x * blockDim.x;
    const uint4 z = make_uint4(0u, 0u, 0u, 0u);
    for (; i < n4; i += stride) ws[i] = z;
}

// ---------------------------------------------------------------------------
// Mark which user / item rows are actually referenced by the batch.
// ---------------------------------------------------------------------------
__global__ void lgcn_mark_kernel(const int* __restrict__ user,
                                 const int* __restrict__ pos,
                                 const int* __restrict__ neg,
                                 unsigned* __restrict__ ubm,
                                 unsigned* __restrict__ ibm, int B) {
    int b = blockIdx.x * blockDim.x + threadIdx.x;
    if (b >= B) return;
    int u = user[b]; atomicOr(&ubm[u >> 5], 1u << (u & 31));
    int p = pos[b];  atomicOr(&ibm[p >> 5], 1u << (p & 31));
    int n = neg[b];  atomicOr(&ibm[n >> 5], 1u << (n & 31));
}

// ---------------------------------------------------------------------------
// Edge scatter, wave-compacted:
//   Phase 1: lane-per-edge -> coalesced loads of 32 edges, dropout + bitmap
//            filter per lane.
//   Phase 2: ballot -> uniform mask of survivors; for each set bit, broadcast
//            (row, col, dval) via v_readlane (scalar row base addresses) and
//            do a full-wave 128-dim gather + f32-atomic scatter (4 dims/lane).
// Only ~15% of edges survive the filter, so compaction cuts wave-iterations
// ~6x vs wave-per-edge, and the edge stream (51 MB from HBM) is coalesced.
// ---------------------------------------------------------------------------
__global__ void lgcn_scatter_kernel(const float* __restrict__ user_emb,
                                    const float* __restrict__ item_emb,
                                    const float* __restrict__ vals,
                                    const float* __restrict__ keep,
                                    const int*  __restrict__ rows,
                                    const int*  __restrict__ cols,
                                    const unsigned* __restrict__ ubm,
                                    const unsigned* __restrict__ ibm,
                                    float* __restrict__ u1,
                                    float* __restrict__ i1, int E) {
    const int lane   = threadIdx.x & 31;
    const int wave   = (blockIdx.x * blockDim.x + threadIdx.x) >> 5;
    const int nwaves = (gridDim.x * blockDim.x) >> 5;
    const int step   = nwaves * 32;

    for (int base = wave * 32; base < E; base += step) {
        const int e = base + lane;

        // Stream prefetch one pass ahead (global_prefetch_b8).
        const int ep = e + step;
        if (ep < E) {
            __builtin_prefetch(&keep[ep], 0, 1);
            __builtin_prefetch(&vals[ep], 0, 1);
            __builtin_prefetch(&rows[ep], 0, 1);
            __builtin_prefetch(&cols[ep], 0, 1);
        }

        float km = 1.0f, v = 0.0f;
        int r = 0, c = 0;
        if (e < E) {
            km = keep[e];
            v  = vals[e];
            r  = rows[e];
            c  = cols[e];
        }
        const float dv   = (km < KEEP_PROB) ? v * INV_KEEP : 0.0f;
        const bool  live = (dv != 0.0f);
        const bool  doI  = live && (((ibm[c >> 5] >> (c & 31)) & 1u) != 0u);
        const bool  doU  = live && (((ubm[r >> 5] >> (r & 31)) & 1u) != 0u);

        // ---- i1[c,:] += dv * user_emb[r,:] for surviving edges ----
        unsigned mI = __builtin_amdgcn_ballot_w32(doI);
        while (mI) {
            const int j = __builtin_ctz(mI);
            mI &= mI - 1u;
            const int   rj  = __builtin_amdgcn_readlane(r, j);
            const int   cj  = __builtin_amdgcn_readlane(c, j);
            const float dvj = bcast_f32(dv, j);
            const float4 us = *(const float4*)(user_emb + (long)rj * DIM + lane * 4);
            float* dst = i1 + (long)cj * DIM + lane * 4;
            unsafeAtomicAdd(dst + 0, dvj * us.x);
            unsafeAtomicAdd(dst + 1, dvj * us.y);
            unsafeAtomicAdd(dst + 2, dvj * us.z);
            unsafeAtomicAdd(dst + 3, dvj * us.w);
        }

        // ---- u1[r,:] += dv * item_emb[c,:] for surviving edges ----
        unsigned mU = __builtin_amdgcn_ballot_w32(doU);
        while (mU) {
            const int j = __builtin_ctz(mU);
            mU &= mU - 1u;
            const int   rj  = __builtin_amdgcn_readlane(r, j);
            const int   cj  = __builtin_amdgcn_readlane(c, j);
            const float dvj = bcast_f32(dv, j);
            const float4 is = *(const float4*)(item_emb + (long)cj * DIM + lane * 4);
            float* dst = u1 + (long)rj * DIM + lane * 4;
            unsafeAtomicAdd(dst + 0, dvj * is.x);
            unsafeAtomicAdd(dst + 1, dvj * is.y);
            unsafeAtomicAdd(dst + 2, dvj * is.z);
            unsafeAtomicAdd(dst + 3, dvj * is.w);
        }
    }
}

// ---------------------------------------------------------------------------
// Scores via v_wmma_f32_16x16x4_f32: per 16-row tile compute U_blk @ I_blk^T
// (K = 128 in 32 k-steps) and extract the diagonal.
// A 16x4 f32 layout: lane L holds M = L&15, K = {kb, kb+1}, kb = (L>>4)*2.
// B 4x16 layout mirrors it with N = L&15. C/D diagonal (M==N==m) sits at
// lane m, acc[m] for m<8 and lane m+16, acc[m-8] for m>=8.
// ---------------------------------------------------------------------------
__global__ void lgcn_score_kernel(const float* __restrict__ user_emb,
                                  const float* __restrict__ item_emb,
                                  const float* __restrict__ u1,
                                  const float* __restrict__ i1,
                                  const int* __restrict__ user,
                                  const int* __restrict__ pos,
                                  const int* __restrict__ neg,
                                  float* __restrict__ out, int B) {
    const int lane   = threadIdx.x & 31;
    const int wave   = (blockIdx.x * blockDim.x + threadIdx.x) >> 5;
    const int nwaves = (gridDim.x * blockDim.x) >> 5;
    const int ntiles = (B + 15) >> 4;

    for (int tile = wave; tile < ntiles; tile += nwaves) {
        const int m  = lane & 15;
        const int kb = (lane >> 4) << 1;          // 0 or 2
        const int b  = min(tile * 16 + m, B - 1); // clamp, keep EXEC full

        const long ur = (long)user[b] * DIM;
        const long pr = (long)pos[b]  * DIM;
        const long nr = (long)neg[b]  * DIM;
        const float* u0p = user_emb + ur; const float* u1p = u1 + ur;
        const float* p0p = item_emb + pr; const float* p1p = i1 + pr;
        const float* n0p = item_emb + nr; const float* n1p = i1 + nr;

        v8f accP = {0.f, 0.f, 0.f, 0.f, 0.f, 0.f, 0.f, 0.f};
        v8f accN = {0.f, 0.f, 0.f, 0.f, 0.f, 0.f, 0.f, 0.f};

#pragma unroll 4
        for (int k0 = 0; k0 < DIM; k0 += 4) {
            const int k = k0 + kb;
            float2 a0 = *(const float2*)(u0p + k);
            float2 a1 = *(const float2*)(u1p + k);
            v2f A;  A.x  = 0.5f * (a0.x + a1.x);  A.y  = 0.5f * (a0.y + a1.y);

            float2 bp0 = *(const float2*)(p0p + k);
            float2 bp1 = *(const float2*)(p1p + k);
            v2f Bp; Bp.x = 0.5f * (bp0.x + bp1.x); Bp.y = 0.5f * (bp0.y + bp1.y);

            float2 bn0 = *(const float2*)(n0p + k);
            float2 bn1 = *(const float2*)(n1p + k);
            v2f Bn; Bn.x = 0.5f * (bn0.x + bn1.x); Bn.y = 0.5f * (bn0.y + bn1.y);

            // (neg_a, A, neg_b, B, c_mod, C, reuse_a, reuse_b)
            accP = __builtin_amdgcn_wmma_f32_16x16x4_f32(
                false, A, false, Bp, (short)0, accP, false, false);
            accN = __builtin_amdgcn_wmma_f32_16x16x4_f32(
                false, A, false, Bn, (short)0, accN, false, false);
        }

        // Diagonal extraction.
        float sp = 0.f, sn = 0.f;
#pragma unroll
        for (int vv = 0; vv < 8; ++vv) {
            if (lane == vv || lane == 24 + vv) { sp = accP[vv]; sn = accN[vv]; }
        }
        const int  row    = tile * 16 + ((lane < 16) ? lane : lane - 16);
        const bool writer = (lane < 8) || (lane >= 24);
        if (writer && row < B) {
            out[row]     = sp;   // pos_score
            out[B + row] = sn;   // neg_score
        }
    }
}

// ---------------------------------------------------------------------------
extern "C" void kernel_launch(void* const* d_in, const int* in_sizes, int n_in,
                              void* d_out, int out_size, void* d_ws, size_t ws_size,
                              hipStream_t stream) {
    const float* user_emb = (const float*)d_in[0];
    const float* item_emb = (const float*)d_in[1];
    const float* vals     = (const float*)d_in[2];
    const float* keep     = (const float*)d_in[3];
    const int*   rows     = (const int*)d_in[4];
    const int*   cols     = (const int*)d_in[5];
    const int*   user     = (const int*)d_in[6];
    const int*   pos      = (const int*)d_in[7];
    const int*   neg      = (const int*)d_in[8];

    const int U = in_sizes[0] / DIM;
    const int I = in_sizes[1] / DIM;
    const int E = in_sizes[2];
    const int B = in_sizes[6];

    // Workspace layout (4-byte words): u1 [U*D] | i1 [I*D] | user bitmap | item bitmap
    float* ws = (float*)d_ws;
    const size_t u1_off  = 0;
    const size_t i1_off  = (size_t)U * DIM;
    const size_t ubm_off = i1_off + (size_t)I * DIM;
    const size_t ubm_w   = ((((size_t)U + 31) / 32) + 3) & ~(size_t)3;
    const size_t ibm_off = ubm_off + ubm_w;
    const size_t ibm_w   = ((((size_t)I + 31) / 32) + 3) & ~(size_t)3;
    const size_t total_w = ibm_off + ibm_w;      // multiple of 4 words

    float*    u1  = ws + u1_off;
    float*    i1  = ws + i1_off;
    unsigned* ubm = (unsigned*)(ws + ubm_off);
    unsigned* ibm = (unsigned*)(ws + ibm_off);

    // 1) Zero accumulators + bitmaps (fresh every call; no cross-call state).
    const long n4 = (long)(total_w / 4);
    lgcn_zero_kernel<<<1024, 256, 0, stream>>>((uint4*)d_ws, n4);

    // 2) Mark batch-referenced rows.
    lgcn_mark_kernel<<<(B + 255) / 256, 256, 0, stream>>>(user, pos, neg, ubm, ibm, B);

    // 3) Wave-compacted, destination-filtered edge scatter.
    lgcn_scatter_kernel<<<2048, 256, 0, stream>>>(user_emb, item_emb, vals, keep,
                                                  rows, cols, ubm, ibm, u1, i1, E);

    // 4) WMMA score tiles (one wave per 16 batch rows).
    const int ntiles = (B + 15) / 16;
    const int blocks = (ntiles * 32 + 255) / 256;
    lgcn_score_kernel<<<blocks > 0 ? blocks : 1, 256, 0, stream>>>(
        user_emb, item_emb, u1, i1, user, pos, neg, (float*)d_out, B);
}